// DeformRoIPooling2D_5669356831616
// MI455X (gfx1250) — compile-verified
//
#include <hip/hip_runtime.h>
#include <hip/hip_bf16.h>
#include <cstdint>
#include <cstddef>

// ---------------------------------------------------------------------------
// Deformable PS-RoI pooling for MI455X (gfx1250, wave32).
//   data  : (4, 256, 64, 64) f32
//   rois  : (N, 5) f32
//   offset: (N, 2, 7, 7) f32
//   out   : (N, 256, 7, 7) f32
// GROUP_SIZE=1, PART_SIZE=OUT_SIZE=7  ->  out channel c reads in channel c,
// part_h=ph, part_w=pw, nc=1 (single class).
// ---------------------------------------------------------------------------

namespace {
constexpr int   kB      = 4;
constexpr int   kC      = 256;
constexpr int   kH      = 64;
constexpr int   kW      = 64;
constexpr int   kHW     = kH * kW;
constexpr int   kP      = 7;
constexpr int   kBins   = kP * kP;   // 49
constexpr float kScale  = 0.0625f;
constexpr float kTStd   = 0.1f;
}

// CDNA5 async global->LDS path (ASYNCcnt), guarded so compilation never breaks.
#if defined(__has_builtin)
#if __has_builtin(__builtin_amdgcn_global_load_async_to_lds_b32) && \
    __has_builtin(__builtin_amdgcn_s_wait_asynccnt)
#define ATHENA_ASYNC_LDS 1
#endif
#endif
#ifndef ATHENA_ASYNC_LDS
#define ATHENA_ASYNC_LDS 0
#endif

#if ATHENA_ASYNC_LDS
typedef __attribute__((address_space(1))) int athena_gint;  // "__device__ int*"
typedef __attribute__((address_space(3))) int athena_lint;  // "__shared__ int*"

__device__ __forceinline__ void athena_async_copy_b32(const float* gsrc,
                                                      float* lds_dst) {
  // AS1 pointer via 64-bit integer round-trip; AS3 pointer via the low 32
  // bits of the generic pointer (== LDS byte offset per the aperture rules).
  __builtin_amdgcn_global_load_async_to_lds_b32(
      (athena_gint*)(uintptr_t)gsrc,
      (athena_lint*)(uint32_t)(uintptr_t)lds_dst,
      0, 0);
}
#endif

struct BinTable {
  float vals[8];     // [0..4]=rois row, [5]=offset_x, [6]=offset_y
  int   toff[16][4]; // per-sample 4 bilinear tap element-offsets
  float w[16][4];    // per-sample 4 bilinear weights (zeroed if invalid)
  int   cnt;         // number of valid samples in this bin
};

__device__ __forceinline__ float bcast_f(float x) {
  return __int_as_float(__builtin_amdgcn_readfirstlane(__float_as_int(x)));
}
__device__ __forceinline__ int bcast_i(int x) {
  return __builtin_amdgcn_readfirstlane(x);
}

// Stage roi scalars into LDS and build the per-bin sample table.
// `mult` = element multiplier applied to (y*W + x): kC for the channel-last
// (transposed) layout, 1 for the original layout.
__device__ __forceinline__ void build_bin_table(
    BinTable& bt, const float* __restrict__ rois, const float* __restrict__ off,
    int n, int r, int ph, int pw, int tid, int mult) {
  if (tid == 0) bt.cnt = 0;
#if ATHENA_ASYNC_LDS
  if (tid < 7) {
    const float* src = (tid < 5)
        ? (rois + (size_t)n * 5 + tid)
        : (off + ((size_t)n * 2 + (tid - 5)) * kBins + r);
    athena_async_copy_b32(src, &bt.vals[tid]);
  }
  __builtin_amdgcn_s_wait_asynccnt(0);
#else
  if (tid < 7) {
    bt.vals[tid] = (tid < 5)
        ? rois[(size_t)n * 5 + tid]
        : off[((size_t)n * 2 + (tid - 5)) * kBins + r];
  }
#endif
  __syncthreads();

  if (tid < 16) {
    const int   iw = tid & 3;
    const int   ih = tid >> 2;
    // rintf == round-half-to-even, matching jnp.round.
    const float x1 = rintf(bt.vals[1]) * kScale - 0.5f;
    const float y1 = rintf(bt.vals[2]) * kScale - 0.5f;
    const float x2 = (rintf(bt.vals[3]) + 1.0f) * kScale - 0.5f;
    const float y2 = (rintf(bt.vals[4]) + 1.0f) * kScale - 0.5f;
    const float roi_w = fmaxf(x2 - x1, 0.1f);
    const float roi_h = fmaxf(y2 - y1, 0.1f);
    const float bin_w = roi_w / (float)kP;
    const float bin_h = roi_h / (float)kP;
    const float sub_w = bin_w / 4.0f;
    const float sub_h = bin_h / 4.0f;
    const float tx = bt.vals[5] * kTStd;
    const float ty = bt.vals[6] * kTStd;
    const float wstart = (float)pw * bin_w + x1 + tx * roi_w;
    const float hstart = (float)ph * bin_h + y1 + ty * roi_h;
    const float wpos = wstart + (float)iw * sub_w;
    const float hpos = hstart + (float)ih * sub_h;
    const bool valid = (wpos >= -0.5f) && (wpos <= (float)kW - 0.5f) &&
                       (hpos >= -0.5f) && (hpos <= (float)kH - 0.5f);
    const float wf = fminf(fmaxf(wpos, 0.0f), (float)(kW - 1));
    const float hf = fminf(fmaxf(hpos, 0.0f), (float)(kH - 1));
    const float xlf = floorf(wf), ylf = floorf(hf);
    const int xl = (int)xlf,        yl = (int)ylf;
    const int xh = (int)ceilf(wf),  yh = (int)ceilf(hf);
    const float dx = wf - xlf;
    const float dy = hf - ylf;
    const float v  = valid ? 1.0f : 0.0f;
    bt.w[tid][0] = (1.0f - dx) * (1.0f - dy) * v;  // (yl, xl)
    bt.w[tid][1] = (1.0f - dx) * dy * v;           // (yh, xl)
    bt.w[tid][2] = dx * (1.0f - dy) * v;           // (yl, xh)
    bt.w[tid][3] = dx * dy * v;                    // (yh, xh)
    bt.toff[tid][0] = (yl * kW + xl) * mult;
    bt.toff[tid][1] = (yh * kW + xl) * mult;
    bt.toff[tid][2] = (yl * kW + xh) * mult;
    bt.toff[tid][3] = (yh * kW + xh) * mult;
    if (valid) atomicAdd(&bt.cnt, 1);
  }
  __syncthreads();
}

// ---------------------------------------------------------------------------
// Layout transform: (B, C, H, W) -> (B, H*W, C), fully coalesced both ways
// via a 64x64 LDS tile (pad 65 -> bank-conflict-free).
// ---------------------------------------------------------------------------
__global__ void __launch_bounds__(256)
transpose_bchw_to_bhwc(const float* __restrict__ in, float* __restrict__ outT) {
  __shared__ float tile[64][65];
  int blk = blockIdx.x;                 // ((b*64 + hwTile)*4 + cTile)
  const int ct  = blk & 3;  blk >>= 2;
  const int hwt = blk & 63; blk >>= 6;
  const int b   = blk;
  const int c0  = ct * 64;
  const int hw0 = hwt * 64;
  const int tid = threadIdx.x;
  const int lx  = tid & 63;             // contiguous within wave
  const int ly  = tid >> 6;             // 0..3
#pragma unroll
  for (int i = 0; i < 16; ++i) {
    const int ch = i * 4 + ly;
    tile[ch][lx] = in[((size_t)b * kC + (c0 + ch)) * kHW + hw0 + lx];
  }
  __syncthreads();
#pragma unroll
  for (int i = 0; i < 16; ++i) {
    const int hw = i * 4 + ly;
    outT[((size_t)b * kHW + (hw0 + hw)) * kC + c0 + lx] = tile[lx][hw];
  }
}

// ---------------------------------------------------------------------------
// Main kernel, channel-last layout: one workgroup (64 threads, 2 waves) per
// (n, ph, pw) bin; each thread covers 4 channels with float4 loads. Tap
// offsets/weights are wave-uniform -> readfirstlane into SGPRs so each tap is
// a single coalesced 512B burst per wave (global_load_b128 with saddr).
// ---------------------------------------------------------------------------
__global__ void __launch_bounds__(64)
deform_roi_pool_t(const float* __restrict__ dataT,
                  const float* __restrict__ rois,
                  const float* __restrict__ off,
                  float* __restrict__ out) {
  __shared__ BinTable bt;
  const int bin = blockIdx.x;
  const int n  = bin / kBins;
  const int r  = bin % kBins;
  const int ph = r / kP;
  const int pw = r % kP;
  const int tid = threadIdx.x;

  build_bin_table(bt, rois, off, n, r, ph, pw, tid, kC);

  const int b = (int)bt.vals[0];
  const float* baseT = dataT + (size_t)b * kHW * kC + tid * 4;

  float4 acc = make_float4(0.0f, 0.0f, 0.0f, 0.0f);
#pragma unroll
  for (int s = 0; s < 16; ++s) {
#pragma unroll
    for (int t = 0; t < 4; ++t) {
      const int   o   = bcast_i(bt.toff[s][t]);
      const float wgt = bcast_f(bt.w[s][t]);
      const float4 v  = *reinterpret_cast<const float4*>(baseT + o);
      acc.x = fmaf(wgt, v.x, acc.x);
      acc.y = fmaf(wgt, v.y, acc.y);
      acc.z = fmaf(wgt, v.z, acc.z);
      acc.w = fmaf(wgt, v.w, acc.w);
    }
  }

  const int   cnt_i = bt.cnt;
  const float cnt   = (float)cnt_i;
  const size_t ob = ((size_t)n * kC + tid * 4) * kBins + r;
  if (cnt_i > 0) {
    out[ob]             = acc.x / cnt;
    out[ob + kBins]     = acc.y / cnt;
    out[ob + 2 * kBins] = acc.z / cnt;
    out[ob + 3 * kBins] = acc.w / cnt;
  } else {
    out[ob] = 0.0f; out[ob + kBins] = 0.0f;
    out[ob + 2 * kBins] = 0.0f; out[ob + 3 * kBins] = 0.0f;
  }
}

// ---------------------------------------------------------------------------
// Fallback (no scratch for the transposed tensor): original layout, one
// thread per channel. Loads are per-lane strided (L2-resident anyway).
// ---------------------------------------------------------------------------
__global__ void __launch_bounds__(256)
deform_roi_pool_direct(const float* __restrict__ data,
                       const float* __restrict__ rois,
                       const float* __restrict__ off,
                       float* __restrict__ out) {
  __shared__ BinTable bt;
  const int bin = blockIdx.x;
  const int n  = bin / kBins;
  const int r  = bin % kBins;
  const int ph = r / kP;
  const int pw = r % kP;
  const int tid = threadIdx.x;

  build_bin_table(bt, rois, off, n, r, ph, pw, tid, 1);

  const int b = (int)bt.vals[0];
  const float* base = data + ((size_t)b * kC + tid) * kHW;

  float acc = 0.0f;
#pragma unroll
  for (int s = 0; s < 16; ++s) {
#pragma unroll
    for (int t = 0; t < 4; ++t) {
      acc = fmaf(bt.w[s][t], base[bt.toff[s][t]], acc);
    }
  }

  const int cnt_i = bt.cnt;
  const size_t ob = ((size_t)n * kC + tid) * kBins + r;
  out[ob] = (cnt_i > 0) ? (acc / (float)cnt_i) : 0.0f;
}

extern "C" void kernel_launch(void* const* d_in, const int* in_sizes, int n_in,
                              void* d_out, int out_size, void* d_ws, size_t ws_size,
                              hipStream_t stream) {
  const float* data = (const float*)d_in[0];
  const float* rois = (const float*)d_in[1];
  const float* off  = (const float*)d_in[2];
  float* out = (float*)d_out;

  const int nrois = in_sizes[1] / 5;
  const int nbins = nrois * kBins;

  const size_t needT = (size_t)kB * kHW * kC * sizeof(float);  // 16 MiB
  if (ws_size >= needT) {
    float* dT = (float*)d_ws;
    transpose_bchw_to_bhwc<<<kB * (kHW / 64) * (kC / 64), 256, 0, stream>>>(data, dT);
    deform_roi_pool_t<<<nbins, 64, 0, stream>>>(dT, rois, off, out);
  } else {
    deform_roi_pool_direct<<<nbins, 256, 0, stream>>>(data, rois, off, out);
  }
}